// Dist_Dense_11802570129553
// MI455X (gfx1250) — compile-verified
//
#include <hip/hip_runtime.h>
#include <stdint.h>

// Tropical (max-plus) dense layer for MI455X / gfx1250:
//   out[b,j] = max_i (x[b,i] - W[i,j]) + bias[j]
// B=128, N=1024, fp32. VALU-bound (268M sub/max ops vs ~5MB traffic; dataset
// fits entirely in the 192MB L2). Strategy:
//   - TDM (tensor_load_to_lds) double-buffered LDS tiles, wave-0 issues,
//     pipelined with s_wait_tensorcnt.
//   - 8-way split of the reduction dim across blockIdx.z for occupancy;
//     partials combined with global_atomic_max_num_f32 (float max is exactly
//     associative -> deterministic).
//   - 4x4 register blocking, 2 i-steps per accumulate so the backend fuses
//     fmax chains into v_max3_num_f32.

#define TB 64    // batch tile per workgroup
#define TJ 64    // output-column tile per workgroup
#define KI 32    // reduction chunk (i) per pipeline stage
#define KSPLIT 8 // reduction split across blockIdx.z
#define NTHREADS 256

typedef __attribute__((ext_vector_type(4))) unsigned int u32x4;
typedef __attribute__((ext_vector_type(8))) int          i32x8;
typedef __attribute__((ext_vector_type(4))) int          i32x4;

#if __has_builtin(__builtin_amdgcn_tensor_load_to_lds)
#define HAVE_TDM 1
#else
#define HAVE_TDM 0
#endif

#if HAVE_TDM
// Issue one 2D TENSOR_LOAD_TO_LDS. Descriptor packing per CDNA5 ISA 8.3/8.4:
//  group0: [1:0]=count(1), [63:32]=lds_addr, [120:64]=global_addr, [127:126]=type(2)
//  group1: [17:16]=data_size(2 -> 4B), [79:48]=tensor_dim0, [111:80]=tensor_dim1,
//          [127:112]=tile_dim0, [143:128]=tile_dim1, [207:160]=tensor_dim0_stride
// groups 2/3 zero (2D tensor, no iterate/gather/multicast).
__device__ __forceinline__ void tdm_load_2d(uint32_t lds_addr, const void* gptr,
                                            uint32_t tensor_d0, uint32_t tensor_d1,
                                            uint32_t tile_d0, uint32_t tile_d1,
                                            uint64_t stride0_elems) {
  const uint64_t ga = (uint64_t)(uintptr_t)gptr;

  u32x4 g0;
  g0.x = 1u;                                               // count=1, user descriptor
  g0.y = lds_addr;                                         // LDS byte address
  g0.z = (uint32_t)ga;                                     // global_addr[31:0]
  g0.w = (uint32_t)((ga >> 32) & 0x1FFFFFFu) | (2u << 30); // addr[56:32] | type=2

  i32x8 g1;
  g1.s0 = (int)(2u << 16);                                          // data_size=4B
  g1.s1 = (int)((tensor_d0 & 0xFFFFu) << 16);                       // dim0[15:0]
  g1.s2 = (int)((tensor_d0 >> 16) | ((tensor_d1 & 0xFFFFu) << 16)); // dim0 hi | dim1 lo
  g1.s3 = (int)((tensor_d1 >> 16) | (tile_d0 << 16));               // dim1 hi | tile0
  g1.s4 = (int)tile_d1;                                             // tile1 (tile2=0)
  g1.s5 = (int)(uint32_t)(stride0_elems & 0xFFFFFFFFull);
  g1.s6 = (int)(uint32_t)(stride0_elems >> 32);
  g1.s7 = 0;

  i32x4 gz = {0, 0, 0, 0};
#if defined(__clang_major__) && (__clang_major__ >= 23)
  i32x8 gz8 = {0, 0, 0, 0, 0, 0, 0, 0};
  __builtin_amdgcn_tensor_load_to_lds(g0, g1, gz, gz, gz8, 0);
#else
  __builtin_amdgcn_tensor_load_to_lds(g0, g1, gz, gz, 0);
#endif
}
#endif  // HAVE_TDM

__global__ __launch_bounds__(NTHREADS) void tropical_init_kernel(float* __restrict__ out,
                                                                 int n) {
  int i = (int)(blockIdx.x * blockDim.x + threadIdx.x);
  if (i < n) out[i] = -__builtin_inff();
}

__global__ __launch_bounds__(NTHREADS) void tropical_dense_kernel(
    const float* __restrict__ x,    // [B][N]
    const float* __restrict__ w,    // [N][N], W[i][j], j contiguous
    const float* __restrict__ bias, // [N]
    float* __restrict__ out,        // [B][N], pre-initialized to -inf
    int N, int B, int ksplit_len) {
  __shared__ float xs[2][TB * KI];  // xs[buf][b][i], i contiguous
  __shared__ float ws[2][KI * TJ];  // ws[buf][i][j], j contiguous

  const int tid = (int)threadIdx.x;
  const int tx = tid & 15;   // j-group: 4 consecutive j
  const int ty = tid >> 4;   // b-group: 4 consecutive b
  const int j0 = (int)blockIdx.x * TJ;
  const int b0 = (int)blockIdx.y * TB;
  const int i0 = (int)blockIdx.z * ksplit_len;  // this WG's reduction range start
  const int K = ksplit_len / KI;                // pipeline stages (4)

  float acc[4][4];
#pragma unroll
  for (int bb = 0; bb < 4; ++bb)
#pragma unroll
    for (int jj = 0; jj < 4; ++jj) acc[bb][jj] = -__builtin_inff();

#if HAVE_TDM
  // One DMA pair per stage: W tile KI x TJ (stride N), x tile TB x KI (stride N).
  auto issue_stage = [&](int k, int buf) {
    tdm_load_2d((uint32_t)(uintptr_t)&ws[buf][0],
                w + (size_t)(i0 + k * KI) * N + j0,
                (uint32_t)N, (uint32_t)N, (uint32_t)TJ, (uint32_t)KI, (uint64_t)N);
    tdm_load_2d((uint32_t)(uintptr_t)&xs[buf][0],
                x + (size_t)b0 * N + (size_t)(i0 + k * KI),
                (uint32_t)N, (uint32_t)B, (uint32_t)KI, (uint32_t)TB, (uint64_t)N);
  };
  if (tid < 32) issue_stage(0, 0);
#endif

  for (int k = 0; k < K; ++k) {
    const int buf = k & 1;

#if HAVE_TDM
    if (tid < 32) {
      if (k + 1 < K) {
        issue_stage(k + 1, buf ^ 1);              // prefetch next stage
        __builtin_amdgcn_s_wait_tensorcnt(2);     // current stage complete
      } else {
        __builtin_amdgcn_s_wait_tensorcnt(0);
      }
    }
    __syncthreads();
#else
    // Fallback: cooperative synchronous tile loads.
    for (int t = tid; t < KI * TJ; t += NTHREADS)
      ws[buf][t] = w[(size_t)(i0 + k * KI + t / TJ) * N + j0 + (t % TJ)];
    for (int t = tid; t < TB * KI; t += NTHREADS)
      xs[buf][t] = x[(size_t)(b0 + t / KI) * N + i0 + k * KI + (t % KI)];
    __syncthreads();
#endif

    const float* xsb = &xs[buf][0];
    const float* wsb = &ws[buf][0];
#pragma unroll 4
    for (int i = 0; i < KI; i += 2) {
      const float4 wa = *(const float4*)(wsb + (i + 0) * TJ + (tx << 2));  // ds_load_b128
      const float4 wb = *(const float4*)(wsb + (i + 1) * TJ + (tx << 2));  // ds_load_b128
      float2 xv[4];
#pragma unroll
      for (int bb = 0; bb < 4; ++bb)
        xv[bb] = *(const float2*)(xsb + (ty * 4 + bb) * KI + i);  // ds_load_b64 broadcast
#pragma unroll
      for (int bb = 0; bb < 4; ++bb) {
        // fmax(fmax(acc, t0), t1) -> v_max3_num_f32 (maxnum composition, exact)
        acc[bb][0] = fmaxf(fmaxf(acc[bb][0], xv[bb].x - wa.x), xv[bb].y - wb.x);
        acc[bb][1] = fmaxf(fmaxf(acc[bb][1], xv[bb].x - wa.y), xv[bb].y - wb.y);
        acc[bb][2] = fmaxf(fmaxf(acc[bb][2], xv[bb].x - wa.z), xv[bb].y - wb.z);
        acc[bb][3] = fmaxf(fmaxf(acc[bb][3], xv[bb].x - wa.w), xv[bb].y - wb.w);
      }
    }
    __syncthreads();
  }

  // Combine partial maxima: out = max(out, partial + bias). Adding bias before
  // the max is exact (bias is constant per j). Float max is associative and
  // commutative -> atomic combine is deterministic.
  const float4 bv = *(const float4*)(bias + j0 + (tx << 2));
#pragma unroll
  for (int bb = 0; bb < 4; ++bb) {
    float* row = out + (size_t)(b0 + ty * 4 + bb) * N + j0 + (tx << 2);
    const float v0 = acc[bb][0] + bv.x;
    const float v1 = acc[bb][1] + bv.y;
    const float v2 = acc[bb][2] + bv.z;
    const float v3 = acc[bb][3] + bv.w;
    __hip_atomic_fetch_max(row + 0, v0, __ATOMIC_RELAXED, __HIP_MEMORY_SCOPE_AGENT);
    __hip_atomic_fetch_max(row + 1, v1, __ATOMIC_RELAXED, __HIP_MEMORY_SCOPE_AGENT);
    __hip_atomic_fetch_max(row + 2, v2, __ATOMIC_RELAXED, __HIP_MEMORY_SCOPE_AGENT);
    __hip_atomic_fetch_max(row + 3, v3, __ATOMIC_RELAXED, __HIP_MEMORY_SCOPE_AGENT);
  }
}

extern "C" void kernel_launch(void* const* d_in, const int* in_sizes, int n_in,
                              void* d_out, int out_size, void* d_ws, size_t ws_size,
                              hipStream_t stream) {
  (void)n_in; (void)d_ws; (void)ws_size;
  const float* x    = (const float*)d_in[0];
  const float* w    = (const float*)d_in[1];
  const float* bias = (const float*)d_in[2];
  float* out        = (float*)d_out;

  const int N = in_sizes[2];          // 1024
  const int B = in_sizes[0] / N;      // 128
  const int ksplit_len = N / KSPLIT;  // 128 i-values per WG

  // Pass 1: out = -inf (required each call; atomic-max combine follows).
  tropical_init_kernel<<<(out_size + NTHREADS - 1) / NTHREADS, NTHREADS, 0, stream>>>(
      out, out_size);

  // Pass 2: tiled max-plus with TDM pipelining; (16, 2, 8) = 256 workgroups.
  dim3 grid((unsigned)(N / TJ), (unsigned)(B / TB), (unsigned)KSPLIT);
  tropical_dense_kernel<<<grid, NTHREADS, 0, stream>>>(x, w, bias, out, N, B,
                                                       ksplit_len);
}